// TransformerLayer_6356551598373
// MI455X (gfx1250) — compile-verified
//
#include <hip/hip_runtime.h>

// ---------------------------------------------------------------------------
// Fused (softmax-free) transformer layer for MI455X (gfx1250), fp32 WMMA path.
//   out = ((X Wqkv^T) -> Q,K,V ; Q (K^T V)) Wo^T + b_o
// Linear attention identity (no softmax): (QK^T)V == Q(K^T V) -> per-head
// K^T V is 128x128; total ~36.6 GFLOP, ~117MB working set (L2-resident).
// Pipeline: double-buffered LDS tiles filled by GLOBAL_LOAD_ASYNC_TO_LDS_B128
// (ASYNCcnt), consumed by V_WMMA_F32_16X16X4_F32. The double-buffer loop is
// unrolled x2 so the buffer index is a compile-time constant (no pointer
// selects / LDS-base mads in the steady state).
// ---------------------------------------------------------------------------

typedef float v2f __attribute__((ext_vector_type(2)));
typedef float v8f __attribute__((ext_vector_type(8)));
// 16-byte chunk type matching the async-to-LDS builtin's parameter.
typedef int v4i_b128 __attribute__((__vector_size__(4 * sizeof(int))));
typedef __attribute__((address_space(1))) v4i_b128* gchunk_p;  // global side
typedef __attribute__((address_space(3))) v4i_b128* schunk_p;  // LDS side

// problem constants (fixed by the reference)
constexpr int BB  = 2;     // batch
constexpr int NN  = 2048;  // tokens
constexpr int DD  = 1024;  // model dim
constexpr int HH  = 8;     // heads
constexpr int DHH = 128;   // head dim

// tiling
constexpr int BM      = 128;  // block tile M
constexpr int BN      = 128;  // block tile N
constexpr int BK      = 16;   // K tile
constexpr int NTHREAD = 256;  // 8 waves (wave32): 4 along M x 2 along N
// LDS pitches (floats). Multiples of 4 -> staged 16B chunks are b128-aligned.
// Pitch 20 -> conflict-free b64 fragment reads; pitch 136 -> the two
// half-wave K offsets hit disjoint bank sets on b32 fragment reads.
constexpr int PADA = 20;   // row-major A staging [BM][BK] pitch
constexpr int PADB = 20;   // row-major B staging [BN][BK] pitch
constexpr int KPT  = 136;  // K-major staging [BK][128] pitch (transposed ops)

#if defined(__HIP_DEVICE_COMPILE__) && \
    __has_builtin(__builtin_amdgcn_global_load_async_to_lds_b128)
#define HAVE_ASYNC_LDS 1
#else
#define HAVE_ASYNC_LDS 0
#endif

// 16B global -> LDS transfer (async on CDNA5 when the builtin exists)
__device__ __forceinline__ void copy16(const float* g, float* l) {
#if HAVE_ASYNC_LDS
  __builtin_amdgcn_global_load_async_to_lds_b128((gchunk_p)g, (schunk_p)l, 0, 0);
#else
  *(float4*)l = *(const float4*)g;  // aligned by construction
#endif
}

__device__ __forceinline__ void wait_copies() {
#if HAVE_ASYNC_LDS
#if __has_builtin(__builtin_amdgcn_s_wait_asynccnt)
  __builtin_amdgcn_s_wait_asynccnt(0);
#else
  asm volatile("s_wait_asynccnt 0x0" ::: "memory");
#endif
#endif
}

// EPI: 0 = plain row-major C (+ optional bias), 1 = QKV head-permute scatter
template <bool TA, bool TB, int EPI>
__global__ __launch_bounds__(NTHREAD) void gemm_f32_wmma(
    const float* __restrict__ A, const float* __restrict__ B,
    const float* __restrict__ bias, float* __restrict__ C,
    int M, int Ncols, int K, int lda, int ldb, int ldc,
    int zMod,
    size_t aOut, size_t aIn, size_t bOut, size_t bIn, size_t cOut, size_t cIn) {
  constexpr int ASZ = TA ? (BK * KPT) : (BM * PADA);
  constexpr int BSZ = TB ? (BN * PADB) : (BK * KPT);
  __shared__ __align__(16) float As[2][ASZ];
  __shared__ __align__(16) float Bs[2][BSZ];

  const int tid    = threadIdx.x;
  const int lane   = tid & 31;
  const int wv     = tid >> 5;
  const int wmBase = (wv & 3) * 32;   // 4 waves along M, 32 rows each
  const int wnBase = (wv >> 2) * 64;  // 2 waves along N, 64 cols each
  const int lane15 = lane & 15;
  const int ksel   = (lane >> 4) << 1;  // lanes 0-15 -> K{0,1}; 16-31 -> K{2,3}

  const int z = blockIdx.z, zo = z / zMod, zi = z % zMod;
  const float* Ag = A + (size_t)zo * aOut + (size_t)zi * aIn;
  const float* Bg = B + (size_t)zo * bOut + (size_t)zi * bIn;
  float*       Cb = C + (size_t)zo * cOut + (size_t)zi * cIn;
  const int m0 = blockIdx.x * BM;
  const int n0 = blockIdx.y * BN;

  // ---- per-thread staging assignment: 2 A chunks + 2 B chunks of 16B ----
  const float *gA0, *gA1, *gB0, *gB1;
  int lA0, lA1, lB0, lB1;
  size_t aStep, bStep;
  if constexpr (!TA) {  // A row-major [M,K]; stage As[m][k], pitch PADA
    const int m_lo = tid >> 2, kq = (tid & 3) << 2;
    gA0 = Ag + (size_t)(m0 + m_lo) * lda + kq;
    gA1 = Ag + (size_t)(m0 + m_lo + 64) * lda + kq;
    lA0 = m_lo * PADA + kq;
    lA1 = (m_lo + 64) * PADA + kq;
    aStep = BK;
  } else {  // A(m,k)=Amem[k*lda+m]; stage As[k][m], pitch KPT
    const int k_lo = tid >> 5, mq = (tid & 31) << 2;
    gA0 = Ag + (size_t)k_lo * lda + m0 + mq;
    gA1 = Ag + (size_t)(k_lo + 8) * lda + m0 + mq;
    lA0 = k_lo * KPT + mq;
    lA1 = (k_lo + 8) * KPT + mq;
    aStep = (size_t)BK * lda;
  }
  if constexpr (TB) {  // B(k,n)=Bmem[n*ldb+k]; stage Bs[n][k], pitch PADB
    const int n_lo = tid >> 2, kq = (tid & 3) << 2;
    gB0 = Bg + (size_t)(n0 + n_lo) * ldb + kq;
    gB1 = Bg + (size_t)(n0 + n_lo + 64) * ldb + kq;
    lB0 = n_lo * PADB + kq;
    lB1 = (n_lo + 64) * PADB + kq;
    bStep = BK;
  } else {  // B row-major [K,N]; stage Bs[k][n], pitch KPT
    const int k_lo = tid >> 5, nq = (tid & 31) << 2;
    gB0 = Bg + (size_t)k_lo * ldb + n0 + nq;
    gB1 = Bg + (size_t)(k_lo + 8) * ldb + n0 + nq;
    lB0 = k_lo * KPT + nq;
    lB1 = (k_lo + 8) * KPT + nq;
    bStep = (size_t)BK * ldb;
  }

  v8f acc[2][4] = {};

  auto stage = [&](int buf) {
    copy16(gA0, &As[buf][lA0]);
    copy16(gA1, &As[buf][lA1]);
    copy16(gB0, &Bs[buf][lB0]);
    copy16(gB1, &Bs[buf][lB1]);
    gA0 += aStep;
    gA1 += aStep;
    gB0 += bStep;
    gB1 += bStep;
  };
  auto fragA = [&](const float* base, int kk, int mi) -> v2f {
    const int m = wmBase + mi * 16 + lane15;
    if constexpr (!TA) {
      return *(const v2f*)&base[m * PADA + kk + ksel];
    } else {
      v2f r;
      r.x = base[(kk + ksel) * KPT + m];
      r.y = base[(kk + ksel + 1) * KPT + m];
      return r;
    }
  };
  auto fragB = [&](const float* base, int kk, int ni) -> v2f {
    const int n = wnBase + ni * 16 + lane15;
    if constexpr (TB) {
      return *(const v2f*)&base[n * PADB + kk + ksel];
    } else {
      v2f r;
      r.x = base[(kk + ksel) * KPT + n];
      r.y = base[(kk + ksel + 1) * KPT + n];
      return r;
    }
  };
  auto compute = [&](int buf) {
    const float* Al = &As[buf][0];
    const float* Bl = &Bs[buf][0];
#pragma unroll
    for (int kk = 0; kk < BK; kk += 4) {
      v2f af[2], bf[4];
      af[0] = fragA(Al, kk, 0);
      af[1] = fragA(Al, kk, 1);
#pragma unroll
      for (int ni = 0; ni < 4; ++ni) bf[ni] = fragB(Bl, kk, ni);
#pragma unroll
      for (int mi = 0; mi < 2; ++mi)
#pragma unroll
        for (int ni = 0; ni < 4; ++ni)
          acc[mi][ni] = __builtin_amdgcn_wmma_f32_16x16x4_f32(
              false, af[mi], false, bf[ni], (short)0, acc[mi][ni], false, false);
    }
  };

  // ---- double-buffered pipeline, unrolled x2 (buf is compile-time const).
  // T = K/BK is even for every launch in this layer (64, 128, 8).
  const int T = K / BK;
  stage(0);
  wait_copies();
  __syncthreads();
  for (int t = 0; t < T - 2; t += 2) {
    stage(1);     // prefetch tile t+1 while computing tile t
    compute(0);
    wait_copies();
    __syncthreads();
    stage(0);     // prefetch tile t+2 while computing tile t+1
    compute(1);
    wait_copies();
    __syncthreads();
  }
  stage(1);       // tile T-1
  compute(0);     // tile T-2
  wait_copies();
  __syncthreads();
  compute(1);     // tile T-1

  // ---- epilogue: C frag layout lanes 0-15: M=v, lanes 16-31: M=v+8 ----
  const int rsel = (lane >> 4) << 3;
#pragma unroll
  for (int mi = 0; mi < 2; ++mi) {
#pragma unroll
    for (int ni = 0; ni < 4; ++ni) {
      const int colG    = n0 + wnBase + ni * 16 + lane15;
      const int rowBase = m0 + wmBase + mi * 16 + rsel;
      if constexpr (EPI == 0) {
        const float bv = (bias != nullptr) ? bias[colG] : 0.0f;
#pragma unroll
        for (int v = 0; v < 8; ++v)
          Cb[(size_t)(rowBase + v) * ldc + colG] = acc[mi][ni][v] + bv;
      } else {
        // scatter into [k, b, h, n, dh]; channel c = (k*DH + d)*H + h
        const int c    = colG;
        const int kidx = c >> 10;  // / (DH*H)
        const int r    = c & 1023;
        const int d    = r >> 3;  // / H
        const int hh   = r & 7;   // % H
        const float bv = bias[c];
#pragma unroll
        for (int v = 0; v < 8; ++v) {
          const int mG = rowBase + v;  // global token row in [0, B*N)
          const int b  = mG >> 11;     // / N
          const int n  = mG & (NN - 1);
          const size_t dst =
              ((((size_t)kidx * BB + b) * HH + hh) * NN + n) * DHH + d;
          Cb[dst] = acc[mi][ni][v] + bv;
        }
      }
    }
  }
}

extern "C" void kernel_launch(void* const* d_in, const int* in_sizes, int n_in,
                              void* d_out, int out_size, void* d_ws, size_t ws_size,
                              hipStream_t stream) {
  (void)in_sizes; (void)n_in; (void)out_size; (void)ws_size;
  const float* x     = (const float*)d_in[0];  // [B,N,D]
  const float* w_qkv = (const float*)d_in[1];  // [3D,D]
  const float* b_qkv = (const float*)d_in[2];  // [3D]
  const float* w_o   = (const float*)d_in[3];  // [D,D]
  const float* b_o   = (const float*)d_in[4];  // [D]
  float* out = (float*)d_out;                  // [B,N,D]

  float* ws = (float*)d_ws;
  const size_t perMat = (size_t)BB * HH * NN * DHH;   // 4,194,304 floats
  float* qkv  = ws;                                   // [3,B,H,N,DH] (50.3 MB)
  float* kvws = qkv + 3 * perMat;                     // [B*H,DH,DH]  (1 MB)
  float* attn = kvws + (size_t)BB * HH * DHH * DHH;   // [B,N,D]      (16.8 MB)

  const float* qb = qkv;
  const float* kb = qkv + 1 * perMat;
  const float* vb = qkv + 2 * perMat;

  dim3 blk(NTHREAD);

  // 1) QKV projection: [4096,1024] @ [3072,1024]^T, scatter+bias epilogue
  gemm_f32_wmma<false, true, 1><<<dim3((BB * NN) / BM, (3 * DD) / BN, 1), blk, 0, stream>>>(
      x, w_qkv, b_qkv, qkv,
      BB * NN, 3 * DD, DD, DD, DD, 0,
      1, 0, 0, 0, 0, 0, 0);

  // 2) KV = K^T V per (b,h): M=128, N=128, K=2048, z = b*H + h
  gemm_f32_wmma<true, false, 0><<<dim3(DHH / BM, DHH / BN, BB * HH), blk, 0, stream>>>(
      kb, vb, nullptr, kvws,
      DHH, DHH, NN, DHH, DHH, DHH,
      1, (size_t)NN * DHH, 0, (size_t)NN * DHH, 0, (size_t)DHH * DHH, 0);

  // 3) attn_out = Q @ KV per (b,h), written directly as [b, n, h*DH+d]
  gemm_f32_wmma<false, false, 0><<<dim3(NN / BM, DHH / BN, BB * HH), blk, 0, stream>>>(
      qb, kvws, nullptr, attn,
      NN, DHH, DHH, DHH, DHH, DD,
      HH,
      (size_t)HH * NN * DHH, (size_t)NN * DHH,    // A: b-stride, h-stride
      (size_t)HH * DHH * DHH, (size_t)DHH * DHH,  // B: b-stride, h-stride
      (size_t)NN * DD, (size_t)DHH);              // C: b-stride, h-stride

  // 4) out = attn @ Wo^T + b_o
  gemm_f32_wmma<false, true, 0><<<dim3((BB * NN) / BM, DD / BN, 1), blk, 0, stream>>>(
      attn, w_o, b_o, out,
      BB * NN, DD, DD, DD, DD, DD,
      1, 0, 0, 0, 0, 0, 0);
}